// GNN_MultiDim_SPATIAL_Block_38328288149511
// MI455X (gfx1250) — compile-verified
//
#include <hip/hip_runtime.h>
#include <hip/hip_bf16.h>

typedef __attribute__((ext_vector_type(2))) float v2f;
typedef __attribute__((ext_vector_type(8))) float v8f;

#define F 64  // F_IN == F_LAT == F_OUT == 64

// ---------------------------------------------------------------------------
// Degree accumulation: integer atomics into deg_out[src], deg_in[dst]
// ---------------------------------------------------------------------------
__global__ void deg_kernel(const int* __restrict__ src, const int* __restrict__ dst,
                           unsigned* __restrict__ deg_out, unsigned* __restrict__ deg_in,
                           int n_edges) {
    int i = blockIdx.x * blockDim.x + threadIdx.x;
    if (i < n_edges) {
        atomicAdd(&deg_out[src[i]], 1u);
        atomicAdd(&deg_in[dst[i]], 1u);
    }
}

// deg (int) -> deg^{-1/2} (0 where deg == 0)
__global__ void isqrt_kernel(const unsigned* __restrict__ deg, float* __restrict__ out, int n) {
    int i = blockIdx.x * blockDim.x + threadIdx.x;
    if (i < n) {
        unsigned d = deg[i];
        out[i] = d ? rsqrtf((float)d) : 0.0f;
    }
}

// ---------------------------------------------------------------------------
// h = (x * dscale[:,None]) @ W   via V_WMMA_F32_16X16X4_F32
// x: [n,64] row-major, W: [64,64] row-major (in,out), h: [n,64]
// One wave -> one 16x16 output tile. Block = 128 thr = 4 waves = the 4 column
// tiles of one 16-row stripe. Grid = n/16 (n is a multiple of 16).
// ---------------------------------------------------------------------------
__global__ __launch_bounds__(128) void gemm_wmma_kernel(const float* __restrict__ x,
                                                        const float* __restrict__ dscale,
                                                        const float* __restrict__ W,
                                                        float* __restrict__ h) {
    const int lane  = threadIdx.x & 31;
    const int tileN = threadIdx.x >> 5;   // 0..3
    const int tileM = blockIdx.x;
    const int l15   = lane & 15;
    const int hi    = lane >> 4;          // 0: lanes 0-15, 1: lanes 16-31
    const int kOff  = hi ? 2 : 0;         // K sub-offset carried by upper half-wave

    const int row = tileM * 16 + l15;     // A-fragment row (M = lane&15 in both halves)
    const int col = tileN * 16 + l15;     // B/C/D column

    const float s = dscale[row];
    const float* __restrict__ xrow = x + (size_t)row * F;

    v8f c = {};
#pragma unroll
    for (int k = 0; k < F; k += 4) {
        v2f a, b;
        // A 16x4 f32 layout: VGPR0 = K=kOff, VGPR1 = K=kOff+1 (per half-wave)
        a.x = xrow[k + kOff]     * s;
        a.y = xrow[k + kOff + 1] * s;
        // B 4x16 f32 layout: row K=kOff / K=kOff+1, column = lane&15
        b.x = W[(size_t)(k + kOff)     * F + col];
        b.y = W[(size_t)(k + kOff + 1) * F + col];
        // 8 args: (neg_a, A, neg_b, B, c_mod, C, reuse_a, reuse_b)
        c = __builtin_amdgcn_wmma_f32_16x16x4_f32(false, a, false, b, (short)0, c,
                                                  false, false);
    }

    // C/D layout: VGPR v -> row (tileM*16 + v + hi*8), col = tileN*16 + (lane&15)
    const int baseRow = tileM * 16 + hi * 8;
#pragma unroll
    for (int v = 0; v < 8; ++v) {
        h[(size_t)(baseRow + v) * F + col] = c[v];
    }
}

// ---------------------------------------------------------------------------
// Edge-parallel scatter-sum: agg[dst[e], 4q..4q+3] += h[src[e], 4q..4q+3]
// One thread per (edge, feature-quad): 16 consecutive threads share an edge ->
// one coalesced global_load_b128 each covers the 64-float source row; the 4
// f32 atomics land in L2 (node features fit in the 192MB L2, so this runs at
// L2 atomic throughput, not HBM).
// ---------------------------------------------------------------------------
__global__ void scatter_kernel(const float* __restrict__ h,
                               const int* __restrict__ src, const int* __restrict__ dst,
                               float* __restrict__ agg, int n_quads) {
    int idx = blockIdx.x * blockDim.x + threadIdx.x;   // E*16 total
    if (idx < n_quads) {
        int e = idx >> 4;
        int q = (idx & 15) << 2;                       // feature offset 0,4,...,60
        const float4 v = *(const float4*)(h + (size_t)src[e] * F + q);
        float* a = agg + (size_t)dst[e] * F + q;
        atomicAdd(a + 0, v.x);
        atomicAdd(a + 1, v.y);
        atomicAdd(a + 2, v.z);
        atomicAdd(a + 3, v.w);
    }
}

// ---------------------------------------------------------------------------
// out = agg * deg_in^{-1/2} + bias  (optional ReLU), float4-vectorized
// ---------------------------------------------------------------------------
__global__ void post_kernel(const float* __restrict__ agg, const float* __restrict__ dii,
                            const float* __restrict__ bias, float* __restrict__ out,
                            int n_quads, int relu) {
    int idx = blockIdx.x * blockDim.x + threadIdx.x;   // N*16 total
    if (idx < n_quads) {
        int node = idx >> 4;
        int q = (idx & 15) << 2;
        const float  d = dii[node];
        const float4 a = *(const float4*)(agg + ((size_t)idx << 2));
        const float4 b = *(const float4*)(bias + q);
        float4 r;
        r.x = a.x * d + b.x;
        r.y = a.y * d + b.y;
        r.z = a.z * d + b.z;
        r.w = a.w * d + b.w;
        if (relu) {
            r.x = fmaxf(r.x, 0.0f);
            r.y = fmaxf(r.y, 0.0f);
            r.z = fmaxf(r.z, 0.0f);
            r.w = fmaxf(r.w, 0.0f);
        }
        *(float4*)(out + ((size_t)idx << 2)) = r;
    }
}

// ---------------------------------------------------------------------------
extern "C" void kernel_launch(void* const* d_in, const int* in_sizes, int n_in,
                              void* d_out, int out_size, void* d_ws, size_t ws_size,
                              hipStream_t stream) {
    const float* x  = (const float*)d_in[0];
    const float* W1 = (const float*)d_in[1];
    const float* b1 = (const float*)d_in[2];
    const float* W2 = (const float*)d_in[3];
    const float* b2 = (const float*)d_in[4];
    const int*  src = (const int*)d_in[5];
    const int*  dst = (const int*)d_in[6];

    const int N = in_sizes[0] / F;      // 50000 (multiple of 16)
    const int E = in_sizes[5];          // 800000
    const int NF = N * F;

    // Workspace layout (256B-aligned slices)
    char* ws = (char*)d_ws;
    size_t off = 0;
    auto alloc = [&](size_t bytes) {
        void* p = ws + off;
        off = (off + bytes + 255) & ~(size_t)255;
        return p;
    };
    unsigned* deg  = (unsigned*)alloc(2 * (size_t)N * sizeof(unsigned)); // deg_out | deg_in
    float*    dois = (float*)alloc((size_t)N * sizeof(float));
    float*    dii  = (float*)alloc((size_t)N * sizeof(float));
    float*    bufA = (float*)alloc((size_t)NF * sizeof(float));  // GEMM output
    float*    bufB = (float*)alloc((size_t)NF * sizeof(float));  // scatter accumulator
    float*    bufC = (float*)alloc((size_t)NF * sizeof(float));  // post-layer-1 activations
    unsigned* deg_out = deg;
    unsigned* deg_in  = deg + N;

    const int TB = 256;
    const int scatter_quads  = E * 16;                 // (edge, feature-quad) pairs
    const int scatter_blocks = (scatter_quads + TB - 1) / TB;
    const int post_quads     = N * 16;
    const int post_blocks    = (post_quads + TB - 1) / TB;

    // Degrees + normalization factors
    hipMemsetAsync(deg, 0, 2 * (size_t)N * sizeof(unsigned), stream);
    deg_kernel<<<(E + TB - 1) / TB, TB, 0, stream>>>(src, dst, deg_out, deg_in, E);
    isqrt_kernel<<<(N + TB - 1) / TB, TB, 0, stream>>>(deg_out, dois, N);
    isqrt_kernel<<<(N + TB - 1) / TB, TB, 0, stream>>>(deg_in,  dii,  N);

    // ---- Layer 1 ----
    gemm_wmma_kernel<<<N / 16, 128, 0, stream>>>(x, dois, W1, bufA);
    hipMemsetAsync(bufB, 0, (size_t)NF * sizeof(float), stream);
    scatter_kernel<<<scatter_blocks, TB, 0, stream>>>(bufA, src, dst, bufB, scatter_quads);
    post_kernel<<<post_blocks, TB, 0, stream>>>(bufB, dii, b1, bufC, post_quads, 1);

    // ---- Layer 2 ----
    gemm_wmma_kernel<<<N / 16, 128, 0, stream>>>(bufC, dois, W2, bufA);
    hipMemsetAsync(bufB, 0, (size_t)NF * sizeof(float), stream);
    scatter_kernel<<<scatter_blocks, TB, 0, stream>>>(bufA, src, dst, bufB, scatter_quads);
    post_kernel<<<post_blocks, TB, 0, stream>>>(bufB, dii, b2, (float*)d_out, post_quads, 0);
}